// Net_7370163880303
// MI455X (gfx1250) — compile-verified
//
#include <hip/hip_runtime.h>
#include <math.h>

// GNN with TransformerConv layers on gfx1250 (MI455X).
// Dense GEMMs -> V_WMMA_F32_16X16X4_F32 (full fp32 matrix pipe, 4-way N-tile
// register blocking). Edge phase -> 3-pass segment softmax with native
// global_atomic_add_f32 into L2-resident buffers.

#define HIDC 64
#define HEADS 4
#define LAYERS 4

typedef float v2f __attribute__((ext_vector_type(2)));
typedef float v8f __attribute__((ext_vector_type(8)));

#define ACT_NONE 0
#define ACT_ELU  1

__device__ __forceinline__ float elu1(float x) { return x > 0.f ? x : expm1f(x); }

// native fp32 global atomic add (global_atomic_add_f32)
__device__ __forceinline__ void atomAddF(float* p, float v) {
  unsafeAtomicAdd(p, v);
}

// monotonic float -> uint encoding for atomicMax-based segment max
__device__ __forceinline__ unsigned f2ord(float f) {
  unsigned u = __float_as_uint(f);
  return (u & 0x80000000u) ? ~u : (u | 0x80000000u);
}
__device__ __forceinline__ float ord2f(unsigned u) {
  return (u & 0x80000000u) ? __uint_as_float(u & 0x7fffffffu) : __uint_as_float(~u);
}

// ---------------------------------------------------------------------------
// Y(Nrows, OUT) = act( X(Nrows, K) @ W(OUT, K)^T + bias )
// One wave -> NT 16x16 output tiles (shared A fragment) via
// V_WMMA_F32_16X16X4_F32. Block = 256 threads = 8 waves = 8 row-tiles.
// f32 WMMA fragment layouts (ISA 7.12.2):
//   A 16x4 : lanes 0-15 hold {K=0,K=1}, lanes 16-31 hold {K=2,K=3}, M=lane%16
//   B 4x16 : lanes 0-15 hold rows {K=0,K=1}, lanes 16-31 rows {K=2,K=3}, N=lane%16
//   C/D    : VGPR j, lanes 0-15 -> M=j, lanes 16-31 -> M=j+8, N=lane%16
// ---------------------------------------------------------------------------
template <int K, int ACT, int NT>
__global__ void gemm_wmma(const float* __restrict__ X, const float* __restrict__ W,
                          const float* __restrict__ bias, float* __restrict__ Y,
                          int Nrows, int OUT) {
  const int lane  = threadIdx.x & 31;
  const int wave  = threadIdx.x >> 5;
  const int mtile = blockIdx.y * 8 + wave;
  const int Mtiles = (Nrows + 15) >> 4;
  if (mtile >= Mtiles) return;                 // wave-uniform exit
  const int laneM = lane & 15;
  const int khalf = lane >> 4;                 // 0: K pair {0,1}; 1: K pair {2,3}
  const int row   = mtile * 16 + laneM;

  const float* __restrict__ xp = X + (size_t)row * K + khalf * 2;

  int col[NT];
  const float* wp[NT];
  v8f acc[NT];
#pragma unroll
  for (int t = 0; t < NT; ++t) {
    col[t] = (blockIdx.x * NT + t) * 16 + laneM;
    const int colL = (col[t] < OUT) ? col[t] : (OUT - 1);  // branchless clamp: EXEC stays all-1
    wp[t] = W + (size_t)colL * K + khalf * 2;
    acc[t] = (v8f){0.f, 0.f, 0.f, 0.f, 0.f, 0.f, 0.f, 0.f};
  }

#pragma unroll
  for (int kk = 0; kk < K; kk += 4) {
    const v2f a = *(const v2f*)(xp + kk);
#pragma unroll
    for (int t = 0; t < NT; ++t) {
      const v2f b = *(const v2f*)(wp[t] + kk);
      acc[t] = __builtin_amdgcn_wmma_f32_16x16x4_f32(false, a, false, b, (short)0,
                                                     acc[t], false, false);
    }
  }

#pragma unroll
  for (int t = 0; t < NT; ++t) {
    if (col[t] < OUT) {
      const float bc = bias[col[t]];
#pragma unroll
      for (int j = 0; j < 8; ++j) {
        const int r = mtile * 16 + j + khalf * 8;
        float val = acc[t][j] + bc;
        if (ACT == ACT_ELU) val = elu1(val);
        Y[(size_t)r * OUT + col[t]] = val;
      }
    }
  }
}

// ---------------------------------------------------------------------------
// Pass 1: alpha[e,h] = (q[dst,h,:] . k[src,h,:]) / 8 ; segment max via atomicMax
// one thread per (edge, head)
// ---------------------------------------------------------------------------
__global__ void edge_score(const float* __restrict__ q, const float* __restrict__ k,
                           const long long* __restrict__ ei, float* __restrict__ alpha,
                           unsigned* __restrict__ amax, int E) {
  const int idx = blockIdx.x * blockDim.x + threadIdx.x;
  if (idx >= E * HEADS) return;
  const int e  = idx >> 2;
  const int hh = idx & 3;
  const int s = (int)ei[e];
  const int d = (int)ei[E + e];
  const float4* __restrict__ qp = (const float4*)(q + (size_t)d * 256 + hh * 64);
  const float4* __restrict__ kp = (const float4*)(k + (size_t)s * 256 + hh * 64);
  float acc = 0.f;
#pragma unroll
  for (int i = 0; i < 16; ++i) {
    const float4 a = qp[i];
    const float4 b = kp[i];
    acc += a.x * b.x + a.y * b.y + a.z * b.z + a.w * b.w;
  }
  acc *= 0.125f;  // 1/sqrt(64)
  alpha[idx] = acc;
  atomicMax(amax + d * HEADS + hh, f2ord(acc));
}

// ---------------------------------------------------------------------------
// Pass 2: ex = exp(alpha - amax[dst]); denom[dst] += ex  (ex kept in-place)
// ---------------------------------------------------------------------------
__global__ void edge_exp(float* __restrict__ alpha, const long long* __restrict__ ei,
                         const unsigned* __restrict__ amax, float* __restrict__ denom,
                         int E) {
  const int idx = blockIdx.x * blockDim.x + threadIdx.x;
  if (idx >= E * HEADS) return;
  const int e  = idx >> 2;
  const int hh = idx & 3;
  const int d = (int)ei[E + e];
  const float ex = expf(alpha[idx] - ord2f(amax[d * HEADS + hh]));
  alpha[idx] = ex;
  atomAddF(denom + d * HEADS + hh, ex);
}

// ---------------------------------------------------------------------------
// Pass 3: agg[dst, h, c] += ex[e,h] * v[src, h, c]   (unnormalized)
// 256-thread block handles 4 edges -> coalesced gather + coalesced f32 atomics
// ---------------------------------------------------------------------------
#define EDGES_PER_BLOCK 4
__global__ void edge_agg(const float* __restrict__ exw, const float* __restrict__ v,
                         const long long* __restrict__ ei, float* __restrict__ agg,
                         int E) {
  const int c = threadIdx.x;                  // 0..255 = head*64 + chan
  const int base = blockIdx.x * EDGES_PER_BLOCK;
#pragma unroll
  for (int i = 0; i < EDGES_PER_BLOCK; ++i) {
    const int e = base + i;
    if (e >= E) return;
    const int s = (int)ei[e];
    const int d = (int)ei[E + e];
    const float w = exw[e * HEADS + (c >> 6)];
    atomAddF(agg + (size_t)d * 256 + c, w * v[(size_t)s * 256 + c]);
  }
}

// ---------------------------------------------------------------------------
// Node update: h = elu( mean_h(agg/denom) + skip )
// ---------------------------------------------------------------------------
__global__ void node_update(const float* __restrict__ agg, const float* __restrict__ denom,
                            const float* __restrict__ skip, float* __restrict__ h, int N) {
  const int idx = blockIdx.x * blockDim.x + threadIdx.x;
  if (idx >= N * HIDC) return;
  const int n = idx >> 6;
  const int c = idx & 63;
  float s = 0.f;
#pragma unroll
  for (int hh = 0; hh < HEADS; ++hh) {
    const float dd = denom[n * HEADS + hh];
    const float aa = agg[(size_t)n * 256 + hh * 64 + c];
    s += (dd > 0.f) ? (aa / dd) : 0.f;
  }
  s *= 0.25f;  // head mean
  h[idx] = elu1(s + skip[idx]);
}

// ---------------------------------------------------------------------------
static void launch_gemm(hipStream_t st, int Kdim, int act, const float* X, const float* W,
                        const float* b, float* Y, int Nrows, int OUT) {
  dim3 blk(256);
  const int Mtiles = (Nrows + 15) / 16;
  const int gy = (Mtiles + 7) / 8;
  if (OUT >= 64) {
    dim3 grd((OUT + 63) / 64, gy);            // NT = 4
    if (Kdim == 8 && act == ACT_ELU)
      gemm_wmma<8, ACT_ELU, 4><<<grd, blk, 0, st>>>(X, W, b, Y, Nrows, OUT);
    else if (Kdim == 64 && act == ACT_ELU)
      gemm_wmma<64, ACT_ELU, 4><<<grd, blk, 0, st>>>(X, W, b, Y, Nrows, OUT);
    else if (Kdim == 64 && act == ACT_NONE)
      gemm_wmma<64, ACT_NONE, 4><<<grd, blk, 0, st>>>(X, W, b, Y, Nrows, OUT);
  } else if (OUT == 32) {
    dim3 grd(1, gy);                          // NT = 2
    if (Kdim == 64 && act == ACT_ELU)
      gemm_wmma<64, ACT_ELU, 2><<<grd, blk, 0, st>>>(X, W, b, Y, Nrows, OUT);
  } else {                                    // OUT == 8, NT = 1
    dim3 grd(1, gy);
    if (Kdim == 32 && act == ACT_NONE)
      gemm_wmma<32, ACT_NONE, 1><<<grd, blk, 0, st>>>(X, W, b, Y, Nrows, OUT);
  }
}

extern "C" void kernel_launch(void* const* d_in, const int* in_sizes, int n_in,
                              void* d_out, int out_size, void* d_ws, size_t ws_size,
                              hipStream_t stream) {
  const float*     x      = (const float*)d_in[0];
  const long long* ei     = (const long long*)d_in[1];
  const float*     enc_w1 = (const float*)d_in[2];
  const float*     enc_b1 = (const float*)d_in[3];
  const float*     enc_w2 = (const float*)d_in[4];
  const float*     enc_b2 = (const float*)d_in[5];
  const float*     Wq     = (const float*)d_in[6];
  const float*     bq     = (const float*)d_in[7];
  const float*     Wk     = (const float*)d_in[8];
  const float*     bk     = (const float*)d_in[9];
  const float*     Wv     = (const float*)d_in[10];
  const float*     bv     = (const float*)d_in[11];
  const float*     Ws     = (const float*)d_in[12];
  const float*     bs     = (const float*)d_in[13];
  const float*     out_w1 = (const float*)d_in[14];
  const float*     out_b1 = (const float*)d_in[15];
  const float*     out_w2 = (const float*)d_in[16];
  const float*     out_b2 = (const float*)d_in[17];
  const float*     out_w3 = (const float*)d_in[18];
  const float*     out_b3 = (const float*)d_in[19];

  const int N = in_sizes[0] / 8;   // 50000
  const int E = in_sizes[1] / 2;   // 500000

  // ---- workspace layout (floats) ----
  float* ws = (float*)d_ws;
  size_t off = 0;
  float*    h     = ws + off; off += (size_t)N * HIDC;        // node features
  float*    q     = ws + off; off += (size_t)N * 256;         // also MLP temp
  float*    k     = ws + off; off += (size_t)N * 256;         // also MLP temp
  float*    v     = ws + off; off += (size_t)N * 256;
  float*    skip  = ws + off; off += (size_t)N * HIDC;
  float*    agg   = ws + off; off += (size_t)N * 256;
  float*    denom = ws + off; off += (size_t)N * HEADS;
  unsigned* amax  = (unsigned*)(ws + off); off += (size_t)N * HEADS;
  float*    ebuf  = ws + off; off += (size_t)E * HEADS;

  const int EH_blocks = (E * HEADS + 255) / 256;
  const int NC_blocks = (N * HIDC + 255) / 256;
  const int EA_blocks = (E + EDGES_PER_BLOCK - 1) / EDGES_PER_BLOCK;

  // ---- encoder: Linear(8,64)+ELU, Linear(64,64)+ELU ----
  launch_gemm(stream, 8,  ACT_ELU, x, enc_w1, enc_b1, q, N, HIDC);   // q as temp
  launch_gemm(stream, 64, ACT_ELU, q, enc_w2, enc_b2, h, N, HIDC);

  // ---- TransformerConv layers ----
  for (int l = 0; l < LAYERS; ++l) {
    const float* Wql = Wq + (size_t)l * 256 * 64;  const float* bql = bq + (size_t)l * 256;
    const float* Wkl = Wk + (size_t)l * 256 * 64;  const float* bkl = bk + (size_t)l * 256;
    const float* Wvl = Wv + (size_t)l * 256 * 64;  const float* bvl = bv + (size_t)l * 256;
    const float* Wsl = Ws + (size_t)l * 64 * 64;   const float* bsl = bs + (size_t)l * 64;

    launch_gemm(stream, 64, ACT_NONE, h, Wql, bql, q,    N, 256);
    launch_gemm(stream, 64, ACT_NONE, h, Wkl, bkl, k,    N, 256);
    launch_gemm(stream, 64, ACT_NONE, h, Wvl, bvl, v,    N, 256);
    launch_gemm(stream, 64, ACT_NONE, h, Wsl, bsl, skip, N, HIDC);

    hipMemsetAsync(agg,   0, (size_t)N * 256 * sizeof(float),      stream);
    hipMemsetAsync(denom, 0, (size_t)N * HEADS * sizeof(float),    stream);
    hipMemsetAsync(amax,  0, (size_t)N * HEADS * sizeof(unsigned), stream); // 0 is below every f2ord value: max identity

    edge_score<<<EH_blocks, 256, 0, stream>>>(q, k, ei, ebuf, amax, E);
    edge_exp  <<<EH_blocks, 256, 0, stream>>>(ebuf, ei, amax, denom, E);
    edge_agg  <<<EA_blocks, 256, 0, stream>>>(ebuf, v, ei, agg, E);
    node_update<<<NC_blocks, 256, 0, stream>>>(agg, denom, skip, h, N);
  }

  // ---- output MLP: 64->64 (ELU) -> 32 (ELU) -> 8 ----
  launch_gemm(stream, 64, ACT_ELU, h, out_w1, out_b1, q, N, 64);
  launch_gemm(stream, 64, ACT_ELU, q, out_w2, out_b2, k, N, 32);
  launch_gemm(stream, 32, ACT_NONE, k, out_w3, out_b3, (float*)d_out, N, 8);
}